// ComplexScaling_2680059592933
// MI455X (gfx1250) — compile-verified
//
#include <hip/hip_runtime.h>
#include <stdint.h>

// Bilinear grid_sample with uniform-scale affine, NHWC [N,1024,1024,2] f32.
// Memory-bound (~512 MB traffic, ~0.6 GFLOP): the win is the data path.
//  - gfx1250 async direct-to-LDS B128 copies stage the two needed input rows
//    per output row (GLOBAL_LOAD_ASYNC_TO_LDS, ASYNCcnt-tracked)
//  - horizontal 4-tap blend served from LDS (ds_load_2addr_b64 pairs,
//    bank-conflict-free for s~1)
//  - coalesced float2 (channel-pair) stores
// H/W are compile-time (fixed by the problem) so row/col math is shifts and
// the 1/W,1/H divides become exact power-of-two multiplies (bit-identical).

static constexpr int kH = 1024;
static constexpr int kW = 1024;

__global__ __launch_bounds__(256) void bilinear_scale_kernel(
    const float* __restrict__ in, const float* __restrict__ theta,
    float* __restrict__ out)
{
    __shared__ float2 srow[2][kW];  // 2 rows x 1024 px x 8 B = 16 KB

    const int tid = threadIdx.x;
    const int row = blockIdx.x & (kH - 1);   // kH is a power of two
    const int n   = blockIdx.x >> 10;

    const float s = 1.0f + theta[0];

    // ---- vertical coordinate for this output row (uniform per block) ----
    constexpr float fH = (float)kH;
    const float yn  = (2.0f * (float)row + 1.0f) * (1.0f / fH) - 1.0f;
    const float iy  = ((s * yn + 1.0f) * fH - 1.0f) * 0.5f;
    const float yc0 = floorf(iy);
    const float yc1 = yc0 + 1.0f;
    const float wy1r = iy - yc0;
    const float wy0r = 1.0f - wy1r;
    const float my0 = (yc0 >= 0.0f && yc0 <= fH - 1.0f) ? 1.0f : 0.0f;
    const float my1 = (yc1 >= 0.0f && yc1 <= fH - 1.0f) ? 1.0f : 0.0f;
    const float wy0 = wy0r * my0;
    const float wy1 = wy1r * my1;
    const int y0 = (int)fminf(fmaxf(yc0, 0.0f), fH - 1.0f);
    const int y1 = (int)fminf(fmaxf(yc1, 0.0f), fH - 1.0f);

    // ---- stage input rows y0, y1 into LDS via async direct-to-LDS B128 ----
    const float2* g0 = (const float2*)in + (((size_t)n << 10) + y0) * kW;
    const float2* g1 = (const float2*)in + (((size_t)n << 10) + y1) * kW;
    constexpr int rowBytes = kW * 8;                       // 8192
    const uint32_t lbase = (uint32_t)(uintptr_t)(&srow[0][0]); // LDS byte offset

    // 256 lanes x 16 B = 4096 B per issue; 2 issues per 8 KB row
#pragma unroll
    for (int k = 0; k < 2; ++k) {
        {
            uint32_t loff = lbase + (uint32_t)(tid * 16 + k * 4096);
            const char* ga = (const char*)g0 + tid * 16 + k * 4096;
            asm volatile("global_load_async_to_lds_b128 %0, %1, off"
                         :: "v"(loff), "v"(ga) : "memory");
        }
        {
            uint32_t loff = lbase + (uint32_t)(rowBytes + tid * 16 + k * 4096);
            const char* ga = (const char*)g1 + tid * 16 + k * 4096;
            asm volatile("global_load_async_to_lds_b128 %0, %1, off"
                         :: "v"(loff), "v"(ga) : "memory");
        }
    }
#if __has_builtin(__builtin_amdgcn_s_wait_asynccnt)
    __builtin_amdgcn_s_wait_asynccnt(0);
#else
    asm volatile("s_wait_asynccnt 0" ::: "memory");
#endif
    __syncthreads();  // all 8 waves' async copies now visible in LDS

    // ---- horizontal 4-tap blend from LDS, 4 columns per thread ----
    float2* orow = (float2*)out + (((size_t)n << 10) + row) * kW;
    constexpr float fW = (float)kW;

#pragma unroll
    for (int k = 0; k < 4; ++k) {
        const int col = tid + k * 256;
        const float xnc = (2.0f * (float)col + 1.0f) * (1.0f / fW) - 1.0f;
        const float ix  = ((s * xnc + 1.0f) * fW - 1.0f) * 0.5f;
        const float xc0 = floorf(ix);
        const float xc1 = xc0 + 1.0f;
        const float wx1r = ix - xc0;
        const float wx0r = 1.0f - wx1r;
        const float mx0 = (xc0 >= 0.0f && xc0 <= fW - 1.0f) ? 1.0f : 0.0f;
        const float mx1 = (xc1 >= 0.0f && xc1 <= fW - 1.0f) ? 1.0f : 0.0f;
        const float wx0 = wx0r * mx0;
        const float wx1 = wx1r * mx1;
        const int x0 = (int)fminf(fmaxf(xc0, 0.0f), fW - 1.0f);
        const int x1 = (int)fminf(fmaxf(xc1, 0.0f), fW - 1.0f);

        const float2 a00 = srow[0][x0];
        const float2 a01 = srow[0][x1];
        const float2 a10 = srow[1][x0];
        const float2 a11 = srow[1][x1];

        float2 r;
        r.x = wy0 * (wx0 * a00.x + wx1 * a01.x) + wy1 * (wx0 * a10.x + wx1 * a11.x);
        r.y = wy0 * (wx0 * a00.y + wx1 * a01.y) + wy1 * (wx0 * a10.y + wx1 * a11.y);
        orow[col] = r;
    }
}

extern "C" void kernel_launch(void* const* d_in, const int* in_sizes, int n_in,
                              void* d_out, int out_size, void* d_ws, size_t ws_size,
                              hipStream_t stream) {
    (void)n_in; (void)out_size; (void)d_ws; (void)ws_size;
    const float* in    = (const float*)d_in[0];
    const float* theta = (const float*)d_in[1];
    float* out = (float*)d_out;

    const int N = in_sizes[0] / (kH * kW * 2);  // 32 for this setup

    dim3 grid((unsigned)(N * kH)), block(256);
    hipLaunchKernelGGL(bilinear_scale_kernel, grid, block, 0, stream,
                       in, theta, out);
}